// KernelNNBoundary_42786464202792
// MI455X (gfx1250) — compile-verified
//
#include <hip/hip_runtime.h>

typedef float v2f __attribute__((ext_vector_type(2)));
typedef float v8f __attribute__((ext_vector_type(8)));

#define WIDTH   32
#define KERW    64
#define KIN     6
#define TILE_E  16
#define NWAVES  4          // waves per block, each owns one 16-edge tile

// LDS layout per wave (floats), padded strides vs. 64 banks
#define HS_STRIDE 33       // h_src tile 16 x 32
#define K_STRIDE  68       // K1 / K2 tiles 16 x 64
#define HS_SZ   (TILE_E * HS_STRIDE)
#define K_SZ    (TILE_E * K_STRIDE)
#define WAVE_LDS (HS_SZ + 2 * K_SZ)

__device__ __forceinline__ v8f wmma_f32_16x16x4(v2f a, v2f b, v8f c) {
  // D = A(16x4) * B(4x16) + C, fp32, wave32
  return __builtin_amdgcn_wmma_f32_16x16x4_f32(false, a, false, b, (short)0, c,
                                               false, false);
}

// -------------------------------------------------------------------------
// Pack K-pairs of a KxN row-major weight so a B fragment (rows k,k+1 at one
// column) is a single contiguous 8-byte load:
//   wp[(p*N + c)*2 + q] = w[(2p+q)*N + c]
// -------------------------------------------------------------------------
__global__ void pack_pairs_kernel(const float* __restrict__ w,
                                  float* __restrict__ wp, int K, int Ncols) {
  int idx = blockIdx.x * blockDim.x + threadIdx.x;
  int tot = (K / 2) * Ncols;
  if (idx < tot) {
    int p = idx / Ncols, c = idx - p * Ncols;
    wp[(size_t)idx * 2 + 0] = w[(size_t)(2 * p) * Ncols + c];
    wp[(size_t)idx * 2 + 1] = w[(size_t)(2 * p + 1) * Ncols + c];
  }
}

// -------------------------------------------------------------------------
// Fused per-edge kernel-MLP (WMMA) + gather + message + atomic scatter.
// One wave handles 16 edges. W[e] (32x32) is never materialized in HBM:
// each C-fragment of K2 @ w3 is contracted with h[src] into 16 registers
// per lane (the C layout makes (edge, out-col) lane-stationary across all
// 64 N-tiles), then flushed with global_atomic_add_f32.
// -------------------------------------------------------------------------
__global__ void __launch_bounds__(32 * NWAVES)
nnconv_msg_kernel(const float* __restrict__ h,
                  const int*   __restrict__ ei, int nE,
                  const float* __restrict__ ea,
                  const float* __restrict__ w1,  const float* __restrict__ b1,
                  const float* __restrict__ w2p, const float* __restrict__ b2,
                  const float* __restrict__ w3p, const float* __restrict__ b3,
                  float* __restrict__ agg) {
  __shared__ float lds[NWAVES][WAVE_LDS];

  const int lane = threadIdx.x & 31;
  const int wave = threadIdx.x >> 5;
  const int half = lane >> 4;      // 0: lanes 0-15, 1: lanes 16-31
  const int sub  = lane & 15;

  float* hs  = &lds[wave][0];                 // [16][HS_STRIDE]
  float* K1s = &lds[wave][HS_SZ];             // [16][K_STRIDE]
  float* K2s = &lds[wave][HS_SZ + K_SZ];      // [16][K_STRIDE]

  const v2f* w2v = (const v2f*)w2p;           // [32 pairs][64 cols]
  const v2f* w3v = (const v2f*)w3p;           // [32 pairs][1024 cols]

  const int tile = blockIdx.x * NWAVES + wave;
  const int e0   = tile * TILE_E;

  // --- stage: this lane is responsible for edge (e0 + sub) ---
  int e   = e0 + sub;
  int ec  = (e < nE) ? e : (nE - 1);          // clamp; invalid edges masked later
  int src = ei[ec];                            // edge_index[0][e]
  int dst = ei[nE + ec];                       // edge_index[1][e]

  float eav[KIN];
#pragma unroll
  for (int i = 0; i < KIN; ++i) eav[i] = ea[ec * KIN + i];

  // gather h[src] tile: 16 edges x 32 features (each lane stores 16 values)
#pragma unroll
  for (int ii = 0; ii < 16; ++ii) {
    int i = half * 16 + ii;
    hs[sub * HS_STRIDE + i] = h[(size_t)src * WIDTH + i];
  }

  // K1 = relu(ea @ w1 + b1): 16x64, K=6 -> plain FMA (32 values per lane)
#pragma unroll
  for (int jj = 0; jj < 32; ++jj) {
    int j = half * 32 + jj;
    float acc = b1[j];
#pragma unroll
    for (int i = 0; i < KIN; ++i) acc = fmaf(eav[i], w1[i * KERW + j], acc);
    K1s[sub * K_STRIDE + j] = fmaxf(acc, 0.0f);
  }

  __syncthreads();

  // --- K2 = relu(K1 @ w2 + b2): M=16, N=64, K=64 via WMMA f32 16x16x4 ---
#pragma unroll
  for (int nt = 0; nt < 4; ++nt) {
    v8f c = {};
#pragma unroll
    for (int kt = 0; kt < 16; ++kt) {
      int ka = kt * 4 + half * 2;       // A layout: lanes0-15 K={0,1}, 16-31 K={2,3}
      v2f a, b;
      a.x = K1s[sub * K_STRIDE + ka];
      a.y = K1s[sub * K_STRIDE + ka + 1];
      b   = w2v[(size_t)(kt * 2 + half) * KERW + nt * 16 + sub];
      c = wmma_f32_16x16x4(a, b, c);
    }
#pragma unroll
    for (int r = 0; r < 8; ++r) {
      int m = r + half * 8;
      int j = nt * 16 + sub;
      K2s[m * K_STRIDE + j] = fmaxf(c[r] + b2[j], 0.0f);
    }
  }

  __syncthreads();

  // --- W = K2 @ w3 + b3 (16 x 1024) fused with msg[m][o] += h_src[m][i]*W ---
  // Lane-stationary accumulators: m in {8*half .. 8*half+7}, o in {sub, 16+sub}
  float msgr[2][8];
#pragma unroll
  for (int r = 0; r < 8; ++r) { msgr[0][r] = 0.0f; msgr[1][r] = 0.0f; }

  for (int i = 0; i < 32; ++i) {               // W row index (input feature)
    float hv[8];
#pragma unroll
    for (int r = 0; r < 8; ++r)
      hv[r] = hs[(r + half * 8) * HS_STRIDE + i];  // broadcast read per half

#pragma unroll
    for (int ob = 0; ob < 2; ++ob) {           // two 16-col N-tiles per row i
      int n = i * 2 + ob;
      v8f c = {};
#pragma unroll
      for (int kt = 0; kt < 16; ++kt) {
        int ka = kt * 4 + half * 2;
        v2f a, b;
        a.x = K2s[sub * K_STRIDE + ka];
        a.y = K2s[sub * K_STRIDE + ka + 1];
        b   = w3v[(size_t)(kt * 2 + half) * 1024 + n * 16 + sub];
        c = wmma_f32_16x16x4(a, b, c);
      }
      float bb = b3[n * 16 + sub];             // == b3[i*32 + ob*16 + sub]
#pragma unroll
      for (int r = 0; r < 8; ++r)
        msgr[ob][r] = fmaf(hv[r], c[r] + bb, msgr[ob][r]);
    }
  }

  // --- scatter: agg[dst[m]][o] += msg[m][o] (register -> global atomics) ---
#pragma unroll
  for (int r = 0; r < 8; ++r) {
    int m = r + half * 8;
    int d = __shfl(dst, m, 32);                // lane m holds dst of edge m
    if (e0 + m < nE) {
      atomicAdd(&agg[(size_t)d * WIDTH + sub],      msgr[0][r]);
      atomicAdd(&agg[(size_t)d * WIDTH + 16 + sub], msgr[1][r]);
    }
  }
}

// -------------------------------------------------------------------------
// h0 = x @ fc1_w + fc1_b ;  h = h0
// -------------------------------------------------------------------------
__global__ void init_h0_kernel(const float* __restrict__ x,
                               const float* __restrict__ fc1w,
                               const float* __restrict__ fc1b,
                               float* __restrict__ h0, float* __restrict__ h,
                               int N) {
  int idx = blockIdx.x * blockDim.x + threadIdx.x;
  if (idx < N * WIDTH) {
    int n = idx >> 5, j = idx & 31;
    float v = fmaf(x[n], fc1w[j], fc1b[j]);
    h0[idx] = v;
    h[idx]  = v;
  }
}

__global__ void count_deg_kernel(const int* __restrict__ ei, int nE,
                                 float* __restrict__ cnt) {
  int e = blockIdx.x * blockDim.x + threadIdx.x;
  if (e < nE) atomicAdd(&cnt[ei[nE + e]], 1.0f);
}

__global__ void invert_cnt_kernel(float* __restrict__ cnt, int len) {
  int i = blockIdx.x * blockDim.x + threadIdx.x;
  if (i < len) cnt[i] = 1.0f / fmaxf(cnt[i], 1.0f);
}

// -------------------------------------------------------------------------
// h_new = relu(agg1*inv1 + agg2*inv2 + h@(root1+root2) + bias1 + bias2) + h0
// one wave per node
// -------------------------------------------------------------------------
__global__ void combine_kernel(const float* __restrict__ h,
                               const float* __restrict__ h0,
                               const float* __restrict__ agg1,
                               const float* __restrict__ agg2,
                               const float* __restrict__ inv1,
                               const float* __restrict__ inv2,
                               const float* __restrict__ root1,
                               const float* __restrict__ bias1,
                               const float* __restrict__ root2,
                               const float* __restrict__ bias2,
                               float* __restrict__ hnew, int N) {
  int gwave = (blockIdx.x * blockDim.x + threadIdx.x) >> 5;
  int lane  = threadIdx.x & 31;
  if (gwave >= N) return;
  size_t base = (size_t)gwave * WIDTH;
  float hv  = h[base + lane];
  float acc = agg1[base + lane] * inv1[gwave] + agg2[base + lane] * inv2[gwave]
            + bias1[lane] + bias2[lane];
#pragma unroll
  for (int i = 0; i < WIDTH; ++i) {
    float hi = __shfl(hv, i, 32);
    acc = fmaf(hi, root1[i * WIDTH + lane] + root2[i * WIDTH + lane], acc);
  }
  hnew[base + lane] = fmaxf(acc, 0.0f) + h0[base + lane];
}

// out[n] = h[n] . fc2_w + fc2_b  (wave per node, shuffle reduce)
__global__ void final_fc2_kernel(const float* __restrict__ h,
                                 const float* __restrict__ fc2w,
                                 const float* __restrict__ fc2b,
                                 float* __restrict__ out, int N) {
  int gwave = (blockIdx.x * blockDim.x + threadIdx.x) >> 5;
  int lane  = threadIdx.x & 31;
  if (gwave >= N) return;
  float v = h[(size_t)gwave * WIDTH + lane] * fc2w[lane];
#pragma unroll
  for (int off = 16; off > 0; off >>= 1) v += __shfl_xor(v, off, 32);
  if (lane == 0) out[gwave] = v + fc2b[0];
}

extern "C" void kernel_launch(void* const* d_in, const int* in_sizes, int n_in,
                              void* d_out, int out_size, void* d_ws, size_t ws_size,
                              hipStream_t stream) {
  const float* x    = (const float*)d_in[0];
  const int*   ei   = (const int*)  d_in[1];
  const float* ea   = (const float*)d_in[2];
  const int*   eib  = (const int*)  d_in[3];
  const float* eab  = (const float*)d_in[4];
  const float* fc1w = (const float*)d_in[5];
  const float* fc1b = (const float*)d_in[6];
  const float* fc2w = (const float*)d_in[7];
  const float* fc2b = (const float*)d_in[8];
  const float* k1w1 = (const float*)d_in[9];
  const float* k1b1 = (const float*)d_in[10];
  const float* k1w2 = (const float*)d_in[11];
  const float* k1b2 = (const float*)d_in[12];
  const float* k1w3 = (const float*)d_in[13];
  const float* k1b3 = (const float*)d_in[14];
  const float* root1 = (const float*)d_in[15];
  const float* bias1 = (const float*)d_in[16];
  const float* k2w1 = (const float*)d_in[17];
  const float* k2b1 = (const float*)d_in[18];
  const float* k2w2 = (const float*)d_in[19];
  const float* k2b2 = (const float*)d_in[20];
  const float* k2w3 = (const float*)d_in[21];
  const float* k2b3 = (const float*)d_in[22];
  const float* root2 = (const float*)d_in[23];
  const float* bias2 = (const float*)d_in[24];

  const int N  = in_sizes[0];
  const int E  = in_sizes[2] / KIN;
  const int EB = in_sizes[4] / KIN;

  float* ws    = (float*)d_ws;
  float* h     = ws;                       // N*32
  float* hnew  = ws + (size_t)N * 32;      // N*32
  float* h0    = ws + (size_t)2 * N * 32;  // N*32
  float* agg1  = ws + (size_t)3 * N * 32;  // N*32
  float* agg2  = ws + (size_t)4 * N * 32;  // N*32 (contiguous with agg1)
  float* inv1  = ws + (size_t)5 * N * 32;  // N
  float* inv2  = inv1 + N;                 // N (contiguous with inv1)
  float* k1w2p = inv2 + N;                 // 64*64
  float* k1w3p = k1w2p + KERW * KERW;      // 64*1024
  float* k2w2p = k1w3p + KERW * 1024;      // 64*64
  float* k2w3p = k2w2p + KERW * KERW;      // 64*1024

  // one-time: pair-packed B layouts so each WMMA B fragment is one b64 load
  pack_pairs_kernel<<<(32 * KERW + 255) / 256, 256, 0, stream>>>(k1w2, k1w2p, KERW, KERW);
  pack_pairs_kernel<<<(32 * 1024 + 255) / 256, 256, 0, stream>>>(k1w3, k1w3p, KERW, 1024);
  pack_pairs_kernel<<<(32 * KERW + 255) / 256, 256, 0, stream>>>(k2w2, k2w2p, KERW, KERW);
  pack_pairs_kernel<<<(32 * 1024 + 255) / 256, 256, 0, stream>>>(k2w3, k2w3p, KERW, 1024);

  // h0 / h init + degree counts (edge structure is depth-invariant)
  hipMemsetAsync(inv1, 0, (size_t)2 * N * sizeof(float), stream);
  init_h0_kernel<<<(N * 32 + 255) / 256, 256, 0, stream>>>(x, fc1w, fc1b, h0, h, N);
  count_deg_kernel<<<(E + 255) / 256, 256, 0, stream>>>(ei, E, inv1);
  count_deg_kernel<<<(EB + 255) / 256, 256, 0, stream>>>(eib, EB, inv2);
  invert_cnt_kernel<<<(2 * N + 255) / 256, 256, 0, stream>>>(inv1, 2 * N);

  const int edges_per_block = TILE_E * NWAVES;
  const int msg_block = 32 * NWAVES;

  for (int layer = 0; layer < 4; ++layer) {
    hipMemsetAsync(agg1, 0, (size_t)2 * N * 32 * sizeof(float), stream);
    nnconv_msg_kernel<<<(E + edges_per_block - 1) / edges_per_block, msg_block, 0, stream>>>(
        h, ei, E, ea, k1w1, k1b1, k1w2p, k1b2, k1w3p, k1b3, agg1);
    nnconv_msg_kernel<<<(EB + edges_per_block - 1) / edges_per_block, msg_block, 0, stream>>>(
        h, eib, EB, eab, k2w1, k2b1, k2w2p, k2b2, k2w3p, k2b3, agg2);
    combine_kernel<<<(N + 7) / 8, 256, 0, stream>>>(
        h, h0, agg1, agg2, inv1, inv2, root1, bias1, root2, bias2, hnew, N);
    float* t = h; h = hnew; hnew = t;     // ping-pong (deterministic: fixed loop)
  }

  final_fc2_kernel<<<(N + 7) / 8, 256, 0, stream>>>(h, fc2w, fc2b, (float*)d_out, N);
}